// PoissonEMModule_54494545052343
// MI455X (gfx1250) — compile-verified
//
#include <hip/hip_runtime.h>
#include <cstdint>
#include <math.h>

// ---- problem geometry (module-level constants in the reference) ----
#define NPHI   190
#define NRAD   223
#define NTS    300
#define N1D    200
#define N2D    200
#define NPIX   (N1D * N2D)      // 40000 pixels, 160000 bytes -> fits CDNA5 320KB LDS
#define NBATCH 2
#define PI_F   3.14159265358979323846f

// ---------------------------------------------------------------------------
// CDNA5 async global->LDS copy (ASYNCcnt-tracked). Per-lane: 16 bytes from the
// lane's global address into the lane's LDS byte address. Flat shared pointers
// truncate to the wave-relative LDS byte address (ISA 10.2 aperture rules).
// ---------------------------------------------------------------------------
__device__ __forceinline__ void async_copy16_to_lds(uint32_t lds_byte_addr,
                                                    const float* gsrc) {
  asm volatile("global_load_async_to_lds_b128 %0, %1, off"
               :
               : "v"(lds_byte_addr), "v"((uint64_t)(uintptr_t)gsrc)
               : "memory");
}

__device__ __forceinline__ void wait_async_zero() {
#if __has_builtin(__builtin_amdgcn_s_wait_asynccnt)
  __builtin_amdgcn_s_wait_asynccnt(0);
#else
  asm volatile("s_wait_asynccnt 0" ::: "memory");
#endif
}

// Ray sample setup shared by forward & adjoint (exact reference geometry).
// Returns initial (u,v) in pixel units for t0 = -299 and per-step (du,dv).
__device__ __forceinline__ void ray_setup(int p, int k, float& u, float& v,
                                          float& du, float& dv) {
  float s, c;
  sincosf((float)p * (PI_F / (float)NPHI), &s, &c);
  float r  = -200.0f + (400.0f / 222.0f) * (float)k;
  float px = r * c + 299.0f * s;   // px = r*cos - t0*sin, t0 = -299
  float py = r * s - 299.0f * c;   // py = r*sin + t0*cos
  u  = (px + 199.0f) * 0.5f;       // (px - ORG)/VOX
  v  = (py + 199.0f) * 0.5f;
  du = -s;                         // d(px)/dt * DT / VOX = -sin
  dv =  c;
}

// Branchless bilinear footprint: clamped indices + OOB-zeroed weights.
// Identical semantics to the reference's np.clip(idx) + inb-masked weights,
// but lowers to v_cndmask/v_min/v_max + unconditional ds accesses (no SALU
// branches in the 300-deep inner loop).
struct Foot {
  int   a00, a01, a10, a11;        // clamped LDS element indices
  float w00, w01, w10, w11;        // masked bilinear weights
};

__device__ __forceinline__ Foot footprint(float u, float v) {
  float fu = floorf(u), fv = floorf(v);
  int   i0 = (int)fu,   j0 = (int)fv;
  float au = u - fu,    av = v - fv;
  float bu = 1.0f - au, bv = 1.0f - av;
  // per-axis masked weights (zero when that corner row/col is out of grid)
  float wi0 = (i0 >= 0  && i0 < N1D)     ? bu : 0.0f;
  float wi1 = (i0 >= -1 && i0 < N1D - 1) ? au : 0.0f;
  float wj0 = (j0 >= 0  && j0 < N2D)     ? bv : 0.0f;
  float wj1 = (j0 >= -1 && j0 < N2D - 1) ? av : 0.0f;
  int i0c = min(max(i0, 0), N1D - 1);
  int i1c = min(max(i0 + 1, 0), N1D - 1);
  int j0c = min(max(j0, 0), N2D - 1);
  int j1c = min(max(j0 + 1, 0), N2D - 1);
  Foot f;
  f.a00 = i0c * N2D + j0c;  f.w00 = wi0 * wj0;
  f.a01 = i0c * N2D + j1c;  f.w01 = wi0 * wj1;
  f.a10 = i1c * N2D + j0c;  f.w10 = wi1 * wj0;
  f.a11 = i1c * N2D + j1c;  f.w11 = wi1 * wj1;
  return f;
}

// ---------------------------------------------------------------------------
// Kernel 1: forward projection + EM ratio.  One workgroup per (phi, batch);
// whole image staged into LDS asynchronously, one rad index per lane.
// ---------------------------------------------------------------------------
__global__ void fwd_ratio_kernel(const float* __restrict__ x,
                                 const float* __restrict__ data,
                                 const float* __restrict__ mult,
                                 const float* __restrict__ addc,
                                 float* __restrict__ ratio) {
  extern __shared__ float limg[];                 // 40000 floats
  const int p = blockIdx.x;
  const int b = blockIdx.y;

  // async-stage x[b] (160 KB) into LDS: 10000 x 16B chunks across 256 lanes
  const float* gimg = x + (size_t)b * NPIX;
  for (int c4 = threadIdx.x; c4 < NPIX / 4; c4 += blockDim.x) {
    uint32_t la = (uint32_t)(uintptr_t)(limg + 4 * c4);
    async_copy16_to_lds(la, gimg + 4 * c4);
  }
  wait_async_zero();
  __syncthreads();

  const int k = threadIdx.x;
  if (k < NRAD) {
    float u, v, du, dv;
    ray_setup(p, k, u, v, du, dv);
    float acc = 0.0f;
#pragma unroll 4
    for (int m = 0; m < NTS; ++m) {
      Foot f = footprint(u, v);
      acc = fmaf(f.w00, limg[f.a00], acc);
      acc = fmaf(f.w01, limg[f.a01], acc);
      acc = fmaf(f.w10, limg[f.a10], acc);
      acc = fmaf(f.w11, limg[f.a11], acc);
      u += du; v += dv;
    }
    size_t idx = ((size_t)b * NPHI + p) * NRAD + k;
    float m_ = mult[idx];
    float e  = fmaf(m_, 2.0f * acc, addc[idx]);   // DT = 2
    ratio[idx] = m_ * data[idx] / e;
  }
}

// ---------------------------------------------------------------------------
// Kernel 2: exact adjoint (same weights, scatter).  Per-workgroup private LDS
// image accumulated with ds_add_f32, flushed with global fp32 atomics.
// ---------------------------------------------------------------------------
__global__ void adj_kernel(const float* __restrict__ ratio,
                           float* __restrict__ acc) {
  extern __shared__ float limg[];                 // 40000 floats
  const int p = blockIdx.x;
  const int b = blockIdx.y;

  for (int pix = threadIdx.x; pix < NPIX; pix += blockDim.x) limg[pix] = 0.0f;
  __syncthreads();

  const int k = threadIdx.x;
  if (k < NRAD) {
    size_t idx = ((size_t)b * NPHI + p) * NRAD + k;
    float rk = ratio[idx] * 2.0f;                 // DT = 2
    float u, v, du, dv;
    ray_setup(p, k, u, v, du, dv);
#pragma unroll 2
    for (int m = 0; m < NTS; ++m) {
      Foot f = footprint(u, v);
      unsafeAtomicAdd(&limg[f.a00], f.w00 * rk);  // ds_add_f32 (no return)
      unsafeAtomicAdd(&limg[f.a01], f.w01 * rk);
      unsafeAtomicAdd(&limg[f.a10], f.w10 * rk);
      unsafeAtomicAdd(&limg[f.a11], f.w11 * rk);
      u += du; v += dv;
    }
  }
  __syncthreads();

  float* gacc = acc + (size_t)b * NPIX;
  for (int pix = threadIdx.x; pix < NPIX; pix += blockDim.x) {
    float val = limg[pix];
    if (val != 0.0f) unsafeAtomicAdd(&gacc[pix], val);   // global_atomic_add_f32
  }
}

// ---------------------------------------------------------------------------
// Kernel 0/3: zero the accumulator; final EM multiplicative update.
// ---------------------------------------------------------------------------
__global__ void zero_kernel(float* __restrict__ acc, int n) {
  int i = blockIdx.x * blockDim.x + threadIdx.x;
  if (i < n) acc[i] = 0.0f;
}

__global__ void final_kernel(const float* __restrict__ x,
                             const float* __restrict__ acc,
                             const float* __restrict__ aones,
                             float* __restrict__ out, int n) {
  int i = blockIdx.x * blockDim.x + threadIdx.x;
  if (i < n) out[i] = x[i] * (acc[i] / aones[i]);
}

// ---------------------------------------------------------------------------
extern "C" void kernel_launch(void* const* d_in, const int* in_sizes, int n_in,
                              void* d_out, int out_size, void* d_ws, size_t ws_size,
                              hipStream_t stream) {
  const float* x     = (const float*)d_in[0];   // (B,1,200,200)
  const float* data  = (const float*)d_in[1];   // (B,190,223)
  const float* mult  = (const float*)d_in[2];   // (B,190,223)
  const float* addc  = (const float*)d_in[3];   // (B,190,223)
  const float* aones = (const float*)d_in[4];   // (B,1,200,200)
  float* out = (float*)d_out;

  float* ws    = (float*)d_ws;
  float* ratio = ws;             // NBATCH*NPHI*NRAD = 84740 floats
  float* acc   = ws + 85000;     // NBATCH*NPIX      = 80000 floats

  const int lds_bytes = NPIX * (int)sizeof(float);   // 160000 B (CDNA5 320KB LDS)
  (void)hipFuncSetAttribute((const void*)fwd_ratio_kernel,
                            hipFuncAttributeMaxDynamicSharedMemorySize, lds_bytes);
  (void)hipFuncSetAttribute((const void*)adj_kernel,
                            hipFuncAttributeMaxDynamicSharedMemorySize, lds_bytes);

  dim3 blk(256);
  int nimg = NBATCH * NPIX;
  zero_kernel<<<(nimg + 255) / 256, blk, 0, stream>>>(acc, nimg);
  fwd_ratio_kernel<<<dim3(NPHI, NBATCH), blk, lds_bytes, stream>>>(x, data, mult,
                                                                   addc, ratio);
  adj_kernel<<<dim3(NPHI, NBATCH), blk, lds_bytes, stream>>>(ratio, acc);
  final_kernel<<<(nimg + 255) / 256, blk, 0, stream>>>(x, acc, aones, out, nimg);
}